// Downsample_PASA_group_softmax_hpf_17652315587406
// MI455X (gfx1250) — compile-verified
//
#include <hip/hip_runtime.h>

typedef __attribute__((ext_vector_type(2))) float v2f;
typedef __attribute__((ext_vector_type(8))) float v8f;

#define KSZ  3
#define K2   9
#define CIN  64
#define HIN  128
#define WIN  128
#define NB   4
#define OH   64
#define OW   64
#define NOC  18           // G*K2 conv output channels
#define NOCP 19           // + one zero pad row for the second WMMA M-tile
#define KF   576          // CIN*K2 contraction length
#define HW   (HIN*WIN)

#define USE_ASYNC_LDS 1   // gfx1250 async global->LDS staging of weights

__device__ __forceinline__ int refl(int i, int n) {
    // torch/jax reflect padding (no edge repeat)
    if (i < 0) i = -i;
    if (i >= n) i = 2 * n - 2 - i;
    return i;
}

// ---------------------------------------------------------------------------
// Kernel 1: strided conv (WMMA f32 16x16x4) + BN + softmax + (ident - soft)
//   grid: NB * 8(rowGroups) * 4(oxTiles) blocks, 256 threads (8 waves)
//   wave w handles output row oy = rowGrp*8 + w, cols ox0..ox0+15
// ---------------------------------------------------------------------------
__global__ __launch_bounds__(256) void pasa_sigma_kernel(
    const float* __restrict__ x, const float* __restrict__ conv_w,
    const float* __restrict__ gamma, const float* __restrict__ beta,
    const float* __restrict__ rmean, const float* __restrict__ rvar,
    float* __restrict__ sig)
{
    __shared__ float sW[NOCP][KF];       // weights reordered: [oc][tap*64 + c]; row 18 = 0
    __shared__ float sLog[8][NOC][16];   // per-wave logit tiles
    __shared__ float sScale[NOC], sShift[NOC];

    const int tid = threadIdx.x;
    const int b = blockIdx.x;
    const int oxTile = b & 3;
    const int rowGrp = (b >> 2) & 7;
    const int n = b >> 5;

    // fold BN into per-channel scale/shift once per block
    if (tid < NOC) {
        const float sc = gamma[tid] * rsqrtf(rvar[tid] + 1e-5f);
        sScale[tid] = sc;
        sShift[tid] = beta[tid] - rmean[tid] * sc;
    }

    // stage weights into LDS with f = tap*64 + c ordering; zero the pad row
    for (int i = tid; i < NOCP * KF; i += 256) {
        const int oc = i / KF;
        const int f = i - oc * KF;
        if (oc < NOC) {
            const int tap = f >> 6;
            const int c = f & 63;
            const float* gp = &conv_w[(oc * CIN + c) * K2 + tap];
#if USE_ASYNC_LDS
            // async DMA: no VGPR round-trip, tracked by ASYNCcnt
            unsigned ldsOff = (unsigned)(unsigned long long)
                (__attribute__((address_space(3))) float*)&sW[oc][f];
            asm volatile("global_load_async_to_lds_b32 %0, %1, off"
                         :: "v"(ldsOff), "v"(gp) : "memory");
#else
            sW[oc][f] = *gp;
#endif
        } else {
            sW[oc][f] = 0.0f;
        }
    }
#if USE_ASYNC_LDS
    asm volatile("s_wait_asynccnt 0x0" ::: "memory");
#endif
    __syncthreads();

    const int wave = tid >> 5;
    const int lane = tid & 31;
    const int pos  = lane & 15;   // WMMA column / output column within tile
    const int kh   = lane >> 4;   // lane-half selects K pair {2kh, 2kh+1}

    const int oy  = rowGrp * 8 + wave;
    const int ox0 = oxTile * 16;
    const int ox  = ox0 + pos;

    const float* xn = x + (size_t)n * CIN * HW;

    v8f acc0 = {};                               // oc rows 0..15
    v8f acc1 = {};                               // oc rows 16..17 (+zeros)
    const int ocA = pos;
    const int ocB = (pos < 2) ? (16 + pos) : NOC;   // clamp to zero pad row 18

    #pragma unroll
    for (int tap = 0; tap < K2; ++tap) {
        const int ti = tap / 3, tj = tap % 3;
        const int rr  = refl(2 * oy + ti - 1, HIN);
        const int cc  = refl(2 * ox + tj - 1, WIN);
        const int off = rr * WIN + cc;          // per-lane, constant over chunk loop
        const int fbase = tap << 6;
        #pragma unroll 4
        for (int cq = 0; cq < 16; ++cq) {
            const int c0 = cq << 2;             // channels c0..c0+3 in this K-chunk
            const int f  = fbase + c0 + 2 * kh;
            const int ck = c0 + 2 * kh;
            v2f bf;
            // B 4x16: lane col = pos, VGPR0/1 = K rows {2kh, 2kh+1} = channels ck, ck+1
            bf.x = xn[(size_t)ck * HW + off];
            bf.y = xn[(size_t)(ck + 1) * HW + off];
            // A 16x4: lane row M, VGPR0/1 = K {2kh, 2kh+1}; 8B-aligned -> ds_load_b64
            const v2f a0 = *(const v2f*)&sW[ocA][f];
            const v2f a1 = *(const v2f*)&sW[ocB][f];
            acc0 = __builtin_amdgcn_wmma_f32_16x16x4_f32(false, a0, false, bf,
                                                         (short)0, acc0, false, false);
            acc1 = __builtin_amdgcn_wmma_f32_16x16x4_f32(false, a1, false, bf,
                                                         (short)0, acc1, false, false);
        }
    }

    // spill logits: C/D layout — lane holds col=pos, rows v + 8*kh
    #pragma unroll
    for (int v = 0; v < 8; ++v)
        sLog[wave][v + 8 * kh][pos] = acc0[v];
    if (kh == 0) {                 // tile1 rows M=0,1 -> oc 16,17
        sLog[wave][16][pos] = acc1[0];
        sLog[wave][17][pos] = acc1[1];
    }
    __syncthreads();

    // BN fold + softmax + high-pass (ident - soft), one lane per position
    if (kh == 0) {
        float lv[NOC];
        float vmax = -1e30f;
        #pragma unroll
        for (int ch = 0; ch < NOC; ++ch) {
            const float t = sLog[wave][ch][pos] * sScale[ch] + sShift[ch];
            lv[ch] = t;
            vmax = fmaxf(vmax, t);
        }
        float ssum = 0.0f;
        #pragma unroll
        for (int ch = 0; ch < NOC; ++ch) { lv[ch] = __expf(lv[ch] - vmax); ssum += lv[ch]; }
        const float inv = 1.0f / ssum;
        #pragma unroll
        for (int ch = 0; ch < NOC; ++ch) {
            const int k = ch % K2;
            const float val = ((k == K2 / 2) ? 1.0f : 0.0f) - lv[ch] * inv;
            sig[(((size_t)n * NOC + ch) * OH + oy) * OW + ox] = val;
        }
    }
}

// ---------------------------------------------------------------------------
// Kernel 2: pixel-adaptive 3x3 weighted sum at strided positions
//   each thread: 1 position x 8 channels (sigma reused across channels)
// ---------------------------------------------------------------------------
__global__ __launch_bounds__(256) void pasa_apply_kernel(
    const float* __restrict__ x, const float* __restrict__ sig,
    float* __restrict__ out)
{
    const int t = blockIdx.x * 256 + threadIdx.x;   // NB*8*OH*OW threads
    const int ox  = t & (OW - 1);
    const int oy  = (t >> 6) & (OH - 1);
    const int cc8 = (t >> 12) & 7;
    const int n   = t >> 15;
    const int c0  = cc8 * 8;
    const int g   = c0 >> 5;                        // group = c/32

    float sg[K2];
    #pragma unroll
    for (int k = 0; k < K2; ++k)
        sg[k] = sig[(((size_t)n * NOC + g * K2 + k) * OH + oy) * OW + ox];

    int rr[3], cl[3];
    #pragma unroll
    for (int d = 0; d < 3; ++d) {
        rr[d] = refl(2 * oy + d - 1, HIN);
        cl[d] = refl(2 * ox + d - 1, WIN);
    }

    #pragma unroll
    for (int ci = 0; ci < 8; ++ci) {
        const int ch = c0 + ci;
        const float* xc = x + ((size_t)n * CIN + ch) * HW;
        float acc = 0.0f;
        #pragma unroll
        for (int ti = 0; ti < 3; ++ti)
            #pragma unroll
            for (int tj = 0; tj < 3; ++tj)
                acc = fmaf(xc[rr[ti] * WIN + cl[tj]], sg[ti * 3 + tj], acc);
        out[(((size_t)n * CIN + ch) * OH + oy) * OW + ox] = acc;
    }
}

extern "C" void kernel_launch(void* const* d_in, const int* in_sizes, int n_in,
                              void* d_out, int out_size, void* d_ws, size_t ws_size,
                              hipStream_t stream) {
    const float* x      = (const float*)d_in[0];
    const float* conv_w = (const float*)d_in[1];
    const float* gamma  = (const float*)d_in[2];
    const float* beta   = (const float*)d_in[3];
    const float* rmean  = (const float*)d_in[4];
    const float* rvar   = (const float*)d_in[5];
    float* out = (float*)d_out;
    float* sig = (float*)d_ws;   // NB*NOC*OH*OW floats = 1,179,648 bytes

    // conv+BN+softmax at strided positions only (4x less conv work)
    pasa_sigma_kernel<<<NB * 8 * 4, 256, 0, stream>>>(x, conv_w, gamma, beta,
                                                      rmean, rvar, sig);
    // weighted 3x3 sum + implicit stride-2 subsample
    pasa_apply_kernel<<<(NB * 8 * OH * OW) / 256, 256, 0, stream>>>(x, sig, out);
}